// CustomReshapeLayer_69681549410663
// MI455X (gfx1250) — compile-verified
//
#include <hip/hip_runtime.h>

// Batched upper-triangle scatter:
//   out[b, i, j] = (j >= i) ? in[b, i*M - i*(i-1)/2 + (j - i)] : 0
// Pure streaming op: 268.5 MB read + 536.9 MB write => ~34.5 us floor @ 23.3 TB/s.
// Gather form writes every output byte exactly once (no zero-fill prepass).
// Non-temporal hints keep the 805 MB stream from thrashing the 192 MB L2.

namespace {
constexpr int       kMat       = 2048;
constexpr long long kTriu      = (long long)kMat * (kMat + 1) / 2;  // 2098176
constexpr int       kBatch     = 32;
constexpr int       kVecPerRow = kMat / 4;                           // 512  = 2^9
constexpr int       kVecPerMat = kMat * kVecPerRow;                  // 2^20
constexpr long long kTotalVec  = (long long)kBatch * kVecPerMat;     // 33554432
constexpr int       kBlock     = 256;                                // 8 wave32s / WGP pass

typedef __attribute__((ext_vector_type(4))) float v4f;
}  // namespace

__global__ __launch_bounds__(kBlock)
void triu_scatter_kernel(const float* __restrict__ in, float* __restrict__ out) {
    // One 16-byte output vector per thread; grid covers the output exactly.
    const unsigned tid = blockIdx.x * kBlock + threadIdx.x;

    const unsigned b   = tid >> 20;                      // tid / kVecPerMat
    const unsigned rem = tid & (kVecPerMat - 1);
    const unsigned i   = rem >> 9;                       // output row
    const unsigned j0  = (rem & (kVecPerRow - 1)) << 2;  // first column of this vec

    v4f v;
    if (j0 + 3u < i) {
        // Entirely in the strict lower triangle -> zeros (just a store).
        v = (v4f){0.0f, 0.0f, 0.0f, 0.0f};
    } else {
        // Packed-row base: element (i, j) lives at rowStart + (j - i),
        // rowStart = i*M - i*(i-1)/2.  Fold the "-i" into the pointer.
        const unsigned long long rowOff =
            (unsigned long long)i * (unsigned)kMat
            - ((unsigned long long)i * (unsigned long long)(i - 1u)) / 2ull
            - (unsigned long long)i;
        const float* __restrict__ p =
            in + (unsigned long long)b * (unsigned long long)kTriu + rowOff + j0;

        if (j0 >= i) {
            // Fully in the upper triangle: 4 consecutive packed elements.
            v.x = __builtin_nontemporal_load(p + 0);
            v.y = __builtin_nontemporal_load(p + 1);
            v.z = __builtin_nontemporal_load(p + 2);
            v.w = __builtin_nontemporal_load(p + 3);
        } else {
            // Vector straddles the diagonal (rare: ~1 vec per row).
            v.x = (j0 + 0u >= i) ? __builtin_nontemporal_load(p + 0) : 0.0f;
            v.y = (j0 + 1u >= i) ? __builtin_nontemporal_load(p + 1) : 0.0f;
            v.z = (j0 + 2u >= i) ? __builtin_nontemporal_load(p + 2) : 0.0f;
            v.w = (j0 + 3u >= i) ? __builtin_nontemporal_load(p + 3) : 0.0f;
        }
    }

    // Coalesced, 16B-aligned, write-once -> global_store_b128 th:NT.
    __builtin_nontemporal_store(v, (v4f*)out + tid);
}

extern "C" void kernel_launch(void* const* d_in, const int* in_sizes, int n_in,
                              void* d_out, int out_size, void* d_ws, size_t ws_size,
                              hipStream_t stream) {
    (void)in_sizes; (void)n_in; (void)d_ws; (void)ws_size; (void)out_size;

    const float* in  = (const float*)d_in[0];
    float*       out = (float*)d_out;

    const unsigned long long nvec   = (unsigned long long)kTotalVec;  // 33,554,432
    const unsigned           blocks = (unsigned)(nvec / kBlock);      // 131,072

    triu_scatter_kernel<<<dim3(blocks), dim3(kBlock), 0, stream>>>(in, out);
}